// Multiheadconf_23716809409048
// MI455X (gfx1250) — compile-verified
//
#include <hip/hip_runtime.h>

typedef __attribute__((ext_vector_type(2))) float v2f;
typedef __attribute__((ext_vector_type(4))) float v4f;
typedef __attribute__((ext_vector_type(8))) float v8f;
typedef __attribute__((ext_vector_type(4))) unsigned int u32x4;
typedef __attribute__((ext_vector_type(4))) int i32x4;
typedef __attribute__((ext_vector_type(8))) int i32x8;

#define B_TOT 16384
#define C_TOT 64
#define F_DIM 128
#define H_DIM 128
#define ROWS_PER_BLOCK 128  // 8 waves * 16 rows
#define CGRP 16             // channels per block (grid.y = C/CGRP)

#if defined(__has_builtin)
#if __has_builtin(__builtin_amdgcn_tensor_load_to_lds) && \
    __has_builtin(__builtin_amdgcn_s_wait_tensorcnt)
#define USE_TDM 1
#endif
#endif
#ifndef USE_TDM
#define USE_TDM 0
#endif

__global__ void zero_out_kernel(float* __restrict__ out, int n) {
  int i = blockIdx.x * blockDim.x + threadIdx.x;
  if (i < n) out[i] = 0.0f;
}

// Pre-swizzle W1 so each wmma-f32 B fragment (K=f,f+1 at column h) is one
// aligned 8-byte pair: Ws[c][ (f/2)*H + h ][2] = { W1[c,f,h], W1[c,f+1,h] }.
// Channel tiles stay contiguous 64 KB blocks -> same 1D TDM descriptor.
__global__ void swizzle_w1_kernel(const float* __restrict__ W1,
                                  float* __restrict__ Ws) {
  int i = blockIdx.x * blockDim.x + threadIdx.x;  // pair idx: C*(F/2)*H total
  const int h     = i & (H_DIM - 1);
  const int fpair = (i >> 7) & (F_DIM / 2 - 1);
  const int c     = i >> 13;
  const float* __restrict__ src =
      W1 + ((size_t)c * F_DIM + 2 * fpair) * H_DIM + h;
  v2f pr;
  pr.x = src[0];
  pr.y = src[H_DIM];
  *(v2f*)(Ws + (size_t)i * 2) = pr;
}

template <bool SWZ>
__global__ __launch_bounds__(256) void flowconf_mlp_wmma(
    const float* __restrict__ Wsrc,  // [C][F*H]: swizzled pairs (SWZ) or raw
    const float* __restrict__ x,     // [B, C, F]
    const float* __restrict__ b1,    // [C, H]
    const float* __restrict__ W2,    // [C, H]
    const float* __restrict__ b2,    // [C]
    float* __restrict__ out) {       // [B]
  // Double-buffered W1_c tiles, DMA'd by the Tensor Data Mover while the
  // other buffer feeds the WMMA pipeline. 128 KB of the WGP's 320 KB LDS.
  __shared__ float sW1[2][F_DIM * H_DIM];

  const int tid    = threadIdx.x;
  const int wave   = tid >> 5;
  const int lane   = tid & 31;
  const int lane16 = lane & 15;
  const int khalf  = lane >> 4;  // 0: K=0,1   1: K=2,3 (ISA f32 A/B layout)

  const int b0 = blockIdx.x * ROWS_PER_BLOCK + wave * 16;
  const int c0 = blockIdx.y * CGRP;

  float b2g = 0.0f;
  for (int c = 0; c < CGRP; ++c) b2g += b2[c0 + c];

  float p[8];
#pragma unroll
  for (int v = 0; v < 8; ++v) p[v] = 0.0f;

#if USE_TDM
  // D#: 1D tile of 16384 f32 (64 KB) from Wsrc_c -> LDS buffer.
  auto tdm_issue = [&](int c, int buf) {
    const unsigned long long ga =
        (unsigned long long)(uintptr_t)(Wsrc + (size_t)c * (F_DIM * H_DIM));
    const unsigned int lds = (unsigned int)(uintptr_t)(&sW1[buf][0]);
    u32x4 g0;
    g0[0] = 1u;                                  // count=1, user descriptor
    g0[1] = lds;                                 // lds_addr
    g0[2] = (unsigned int)ga;                    // global_addr[31:0]
    g0[3] = (unsigned int)((ga >> 32) & 0x01FFFFFFu) | (2u << 30);  // |type=2
    i32x8 g1;
    g1[0] = 0x00020000;           // workgroup_mask=0, data_size=2 (4B)
    g1[1] = (int)(16384u << 16);  // tensor_dim0[15:0]=16384
    g1[2] = 0x00010000;           // tensor_dim0 hi=0, tensor_dim1=1
    g1[3] = (int)(16384u << 16);  // tensor_dim1 hi=0, tile_dim0=16384
    g1[4] = 0;                    // tile_dim1=0, tile_dim2=0
    g1[5] = 16384;                // tensor_dim0_stride[31:0]
    g1[6] = 0;
    g1[7] = 0;
    i32x4 gz = {0, 0, 0, 0};
#if __clang_major__ >= 23
    i32x8 gz8 = {0, 0, 0, 0, 0, 0, 0, 0};
    __builtin_amdgcn_tensor_load_to_lds(g0, g1, gz, gz, gz8, 0);
#else
    __builtin_amdgcn_tensor_load_to_lds(g0, g1, gz, gz, 0);
#endif
  };
  if (wave == 0) tdm_issue(c0, 0);  // prologue: prefetch first channel
#endif

  for (int cc = 0; cc < CGRP; ++cc) {
    const int c = c0 + cc;
    const float* __restrict__ sbuf = sW1[cc & 1];

#if USE_TDM
    if (wave == 0) __builtin_amdgcn_s_wait_tensorcnt(0);
    __syncthreads();  // buf[cc&1] ready; all waves done reading buf[(cc+1)&1]
    if (wave == 0 && cc + 1 < CGRP) tdm_issue(c + 1, (cc + 1) & 1);
#else
    __syncthreads();
    {
      const float* __restrict__ Wc = Wsrc + (size_t)c * (F_DIM * H_DIM);
      float* __restrict__ dst = sW1[cc & 1];
      for (int i = tid; i < (F_DIM * H_DIM) / 4; i += 256)
        ((v4f*)dst)[i] = ((const v4f*)Wc)[i];
    }
    __syncthreads();
#endif

    // A fragments: 16 rows of x[., c, :], f32 16x16x4 layout.
    // lane L: row = b0 + L%16, VGPR pair = x[row, 4k + 2*khalf + {0,1}].
    // x is streamed exactly once -> non-temporal.
    v2f a[32];
    {
      const v2f* __restrict__ xrow = (const v2f*)(
          x + ((size_t)(b0 + lane16) * C_TOT + c) * F_DIM + 2 * khalf);
#pragma unroll
      for (int kk = 0; kk < 32; ++kk)
        a[kk] = __builtin_nontemporal_load(xrow + kk * 2);
    }

    const float* __restrict__ b1c = b1 + c * H_DIM;
    const float* __restrict__ W2c = W2 + c * H_DIM;

#pragma unroll
    for (int nt = 0; nt < 8; ++nt) {
      const int h = nt * 16 + lane16;  // this lane's N column
      v8f acc = {};
      if constexpr (SWZ) {
        // pair index = (2kk+khalf)*128 + h -> one aligned b64 per fragment
        const v2f* __restrict__ bp = (const v2f*)sbuf + khalf * H_DIM + h;
#pragma unroll
        for (int kk = 0; kk < 32; ++kk) {
          v2f bf = bp[kk * 256];
          acc = __builtin_amdgcn_wmma_f32_16x16x4_f32(
              false, a[kk], false, bf, (short)0, acc, false, false);
        }
      } else {
        const float* __restrict__ bp = sbuf + khalf * 2 * H_DIM + h;
#pragma unroll
        for (int kk = 0; kk < 32; ++kk) {
          v2f bf;
          bf.x = bp[kk * 4 * H_DIM];
          bf.y = bp[kk * 4 * H_DIM + H_DIM];
          acc = __builtin_amdgcn_wmma_f32_16x16x4_f32(
              false, a[kk], false, bf, (short)0, acc, false, false);
        }
      }
      // fused epilogue: +b1, ReLU, * W2 -> per-lane row partials.
      // C/D layout: VGPR v holds (M = v + 8*khalf, N = lane%16).
      const float b1v = b1c[h];
      const float w2v = W2c[h];
#pragma unroll
      for (int v = 0; v < 8; ++v) {
        float hv = acc[v] + b1v;
        hv = fmaxf(hv, 0.0f);
        p[v] += hv * w2v;
      }
    }
  }

  // Reduce partials over N: sum across the 16 lanes of each half-wave.
#pragma unroll
  for (int v = 0; v < 8; ++v) {
    float s = p[v];
    s += __shfl_xor(s, 1, 32);
    s += __shfl_xor(s, 2, 32);
    s += __shfl_xor(s, 4, 32);
    s += __shfl_xor(s, 8, 32);
    p[v] = s;
  }
  if (lane16 == 0) {
    const int rbase = b0 + khalf * 8;  // lane 0: rows M0..7, lane 16: M8..15
#pragma unroll
    for (int v = 0; v < 8; ++v)
      atomicAdd(&out[rbase + v], p[v] + b2g);
  }
}

extern "C" void kernel_launch(void* const* d_in, const int* in_sizes, int n_in,
                              void* d_out, int out_size, void* d_ws, size_t ws_size,
                              hipStream_t stream) {
  (void)in_sizes; (void)n_in;
  const float* x  = (const float*)d_in[0];  // [B,C,F]
  const float* W1 = (const float*)d_in[1];  // [C,F,H]
  const float* b1 = (const float*)d_in[2];  // [C,H]
  const float* W2 = (const float*)d_in[3];  // [C,H]
  const float* b2 = (const float*)d_in[4];  // [C]
  float* out = (float*)d_out;               // [B]

  zero_out_kernel<<<(out_size + 255) / 256, 256, 0, stream>>>(out, out_size);

  dim3 grid(B_TOT / ROWS_PER_BLOCK, C_TOT / CGRP);
  const size_t w1_bytes = (size_t)C_TOT * F_DIM * H_DIM * sizeof(float);
  if (ws_size >= w1_bytes) {
    float* Ws = (float*)d_ws;
    const int npairs = C_TOT * (F_DIM / 2) * H_DIM;
    swizzle_w1_kernel<<<npairs / 256, 256, 0, stream>>>(W1, Ws);
    flowconf_mlp_wmma<true><<<grid, 256, 0, stream>>>(Ws, x, b1, W2, b2, out);
  } else {
    flowconf_mlp_wmma<false><<<grid, 256, 0, stream>>>(W1, x, b1, W2, b2, out);
  }
}